// AttentionPooling_58368605552826
// MI455X (gfx1250) — compile-verified
//
#include <hip/hip_runtime.h>
#include <hip/hip_bf16.h>
#include <math.h>

// Problem constants (from reference)
#define BDIM 8
#define NDIM 4096
#define CDIM 1024
#define ATTD 512
#define MDIM 64

typedef __attribute__((ext_vector_type(2))) float v2f;
typedef __attribute__((ext_vector_type(8))) float v8f;
typedef __attribute__((ext_vector_type(4))) int v4i;

// gfx1250 async global->LDS copies (ASYNCcnt-tracked), if toolchain has them.
#if defined(__has_builtin)
#if __has_builtin(__builtin_amdgcn_global_load_async_to_lds_b128) && \
    __has_builtin(__builtin_amdgcn_global_load_async_to_lds_b32) &&  \
    __has_builtin(__builtin_amdgcn_s_wait_asynccnt)
#define USE_ASYNC_LDS 1
#endif
#endif
#ifndef USE_ASYNC_LDS
#define USE_ASYNC_LDS 0
#endif

// ---------------------------------------------------------------------------
// Phase 0: fold attention_vectors into the projection, K-pair-packed:
//   Vt2[c/2][m] = { sum_d av[m,d]*W[d,c_even],  sum_d av[m,d]*W[d,c_even+1] }
//   beta[m]     = sum_d av[m,d] * bias[d]
// Pair packing makes each WMMA B-fragment a single b64 load later.
// ---------------------------------------------------------------------------
__global__ __launch_bounds__(256) void ap_fold_kernel(
    const float* __restrict__ W, const float* __restrict__ bias,
    const float* __restrict__ av, float* __restrict__ Vt2,
    float* __restrict__ beta) {
  const int flat = blockIdx.x * 256 + threadIdx.x;  // 0 .. 65535
  const int m = flat & (MDIM - 1);
  const int c = flat >> 6;  // 0 .. 1023
  const float* avrow = av + m * ATTD;
  float s = 0.f;
#pragma unroll 8
  for (int d = 0; d < ATTD; ++d) s += avrow[d] * W[(size_t)d * CDIM + c];
  Vt2[((size_t)(c >> 1) * MDIM + m) * 2 + (c & 1)] = s;
  if (flat < MDIM) {
    float t = 0.f;
#pragma unroll 8
    for (int d = 0; d < ATTD; ++d) t += av[flat * ATTD + d] * bias[d];
    beta[flat] = t;
  }
}

// ---------------------------------------------------------------------------
// Phase 1: logits (pair-packed, n-major):
//   logitsP[b][n/2][m] = { logit(b,n_even,m), logit(b,n_even+1,m) }
// One block = 16 n-rows; 4 waves = the 4 m-tiles of 16.
// x staged in LDS in 512-wide K chunks (stride 516 -> conflict-free
// ds_load_b64 A fragments). WMMA: D(16n x 16m) += A(16n x 4c) * B(4c x 16m).
// ---------------------------------------------------------------------------
#define KCHUNK 512
#define LDSTRIDE 516  // 512 + 4 floats pad: lane stride 4 banks

__global__ __launch_bounds__(128) void ap_logits_kernel(
    const float* __restrict__ x, const float2* __restrict__ Vt2,
    const float* __restrict__ beta, float2* __restrict__ logitsP) {
  __shared__ __align__(16) float xs[16 * LDSTRIDE];

  const int b = blockIdx.y;
  const int n0 = blockIdx.x * 16;
  const int tid = (int)threadIdx.x;
  const int lane = tid & 31;
  const int wave = tid >> 5;
  const int m0 = wave * 16;
  const int lane15 = lane & 15;
  const int khalf = (lane >> 4) * 2;  // 0 for lanes 0-15, 2 for lanes 16-31

  const float* xrow = x + ((size_t)b * NDIM + n0) * CDIM;

  v8f acc = {};

  for (int kc = 0; kc < CDIM; kc += KCHUNK) {
    __syncthreads();  // protect previous chunk's LDS reads
    // Stage 16 rows x 512 floats; iteration i loads row i fully (coalesced).
#pragma unroll 4
    for (int i = 0; i < 16; ++i) {
#if USE_ASYNC_LDS
      __builtin_amdgcn_global_load_async_to_lds_b128(
          (v4i*)(xrow + (size_t)i * CDIM + kc + tid * 4),
          (v4i*)&xs[i * LDSTRIDE + tid * 4], 0, 0);
#else
      const float4 v = *(const float4*)(xrow + (size_t)i * CDIM + kc + tid * 4);
      ((float4*)&xs[i * LDSTRIDE])[tid] = v;
#endif
      if (kc + KCHUNK < CDIM)  // warm caches for next chunk
        __builtin_prefetch(xrow + (size_t)i * CDIM + kc + KCHUNK + tid * 4, 0, 1);
    }
#if USE_ASYNC_LDS
    __builtin_amdgcn_s_wait_asynccnt(0);
#endif
    __syncthreads();

    const float2* vtb = Vt2 + (size_t)(kc >> 1) * MDIM + m0 + lane15;
#pragma unroll 4
    for (int kk = 0; kk < KCHUNK; kk += 4) {
      const int kp = kk + khalf;  // always even
      // A fragment: row n = lane15, K = kp, kp+1 (8B-aligned LDS b64)
      v2f a = *(const v2f*)&xs[lane15 * LDSTRIDE + kp];
      // B fragment: one b64, coalesced across lanes
      const float2 bv = vtb[(size_t)(kp >> 1) * MDIM];
      v2f bb;
      bb.x = bv.x;
      bb.y = bv.y;
      acc = __builtin_amdgcn_wmma_f32_16x16x4_f32(
          false, a, false, bb, (short)0, acc, false, false);
    }
  }

  // D: VGPR v, lanes 0-15 -> row(n)=v, col(m)=lane; lanes 16-31 -> row v+8.
  // Pack consecutive-n pairs -> 4x global_store_b64, coalesced in m.
  const float bet = beta[m0 + lane15];
  const int nbase = n0 + ((lane >> 4) << 3);  // even
  float2* ob = logitsP + (size_t)b * (NDIM / 2) * MDIM +
               (size_t)(nbase >> 1) * MDIM + m0 + lane15;
#pragma unroll
  for (int v2 = 0; v2 < 4; ++v2) {
    float2 val;
    val.x = acc[2 * v2] + bet;
    val.y = acc[2 * v2 + 1] + bet;
    ob[(size_t)v2 * MDIM] = val;
  }
}

// ---------------------------------------------------------------------------
// Phase 2: softmax over n (4096) per (b,m), in-place on pair-packed logits.
// One block per b; thread = (m = tid&63, r = tid>>6). float2 reads, coalesced.
// ---------------------------------------------------------------------------
__global__ __launch_bounds__(512) void ap_softmax_kernel(
    float2* __restrict__ logitsP) {
  __shared__ float red[8][MDIM];
  __shared__ float stat[MDIM];
  const int b = blockIdx.x;
  const int tid = (int)threadIdx.x;
  const int m = tid & (MDIM - 1);
  const int r = tid >> 6;  // 0..7
  float2* base = logitsP + (size_t)b * (NDIM / 2) * MDIM + m;

  float mx = -INFINITY;
  for (int p = r; p < NDIM / 2; p += 8) {
    const float2 v = base[(size_t)p * MDIM];
    mx = fmaxf(mx, fmaxf(v.x, v.y));
  }
  red[r][m] = mx;
  __syncthreads();
  if (r == 0) {
    float g = red[0][m];
#pragma unroll
    for (int j = 1; j < 8; ++j) g = fmaxf(g, red[j][m]);
    stat[m] = g;
  }
  __syncthreads();
  const float g = stat[m];

  float s = 0.f;
  for (int p = r; p < NDIM / 2; p += 8) {
    float2 v = base[(size_t)p * MDIM];
    v.x = __expf(v.x - g);
    v.y = __expf(v.y - g);
    base[(size_t)p * MDIM] = v;
    s += v.x + v.y;
  }
  __syncthreads();
  red[r][m] = s;
  __syncthreads();
  if (r == 0) {
    float t = 0.f;
#pragma unroll
    for (int j = 0; j < 8; ++j) t += red[j][m];
    stat[m] = 1.0f / t;
  }
  __syncthreads();
  const float inv = stat[m];
  for (int p = r; p < NDIM / 2; p += 8) {
    float2 v = base[(size_t)p * MDIM];
    v.x *= inv;
    v.y *= inv;
    base[(size_t)p * MDIM] = v;
  }
}

// ---------------------------------------------------------------------------
// Phase 3: out[b,m,c] = sum_n w[b,n,m] * x[b,n,c]
// Block = one 16-wide c tile; 4 waves = 4 m-tiles sharing the x chunk in LDS
// (4x reuse). WMMA: D(16m x 16c) += A(16m x 4n) * B(4n x 16c).
// A fragments are single b64 loads thanks to the pair-packed weights.
// ---------------------------------------------------------------------------
#define NCHUNK 128

__global__ __launch_bounds__(128) void ap_pool_kernel(
    const float* __restrict__ x, const float2* __restrict__ wP,
    float* __restrict__ out) {
  __shared__ __align__(16) float xs[NCHUNK * 16];

  const int b = blockIdx.y;
  const int c0 = blockIdx.x * 16;
  const int tid = (int)threadIdx.x;
  const int lane = tid & 31;
  const int wave = tid >> 5;
  const int m0 = wave * 16;
  const int lane15 = lane & 15;
  const int khalf = (lane >> 4) * 2;

  const float* xb = x + (size_t)b * NDIM * CDIM + c0;
  const float2* wb = wP + (size_t)b * (NDIM / 2) * MDIM + m0 + lane15;

  v8f acc = {};

  for (int nc = 0; nc < NDIM; nc += NCHUNK) {
    __syncthreads();
    // Stage 128 rows x 16 cols: iteration i covers rows i*8 .. i*8+7,
    // thread t -> row i*8 + (t>>4), col t&15 (64B coalesced per 16 threads).
#pragma unroll 4
    for (int i = 0; i < 16; ++i) {
#if USE_ASYNC_LDS
      __builtin_amdgcn_global_load_async_to_lds_b32(
          (int*)(xb + (size_t)(nc + i * 8 + (tid >> 4)) * CDIM + (tid & 15)),
          (int*)&xs[i * 128 + tid], 0, 0);
#else
      xs[i * 128 + tid] =
          xb[(size_t)(nc + i * 8 + (tid >> 4)) * CDIM + (tid & 15)];
#endif
      if (nc + NCHUNK < NDIM)
        __builtin_prefetch(
            xb + (size_t)(nc + NCHUNK + i * 8 + (tid >> 4)) * CDIM + (tid & 15),
            0, 1);
    }
#if USE_ASYNC_LDS
    __builtin_amdgcn_s_wait_asynccnt(0);
#endif
    __syncthreads();

#pragma unroll 4
    for (int kk = 0; kk < NCHUNK; kk += 4) {
      const int kp = kk + khalf;  // always even
      // A fragment: A[m][k] = w[b][nc+kp .. +1][m0+lane15] (one b64)
      const float2 av2 = wb[(size_t)((nc + kp) >> 1) * MDIM];
      v2f a;
      a.x = av2.x;
      a.y = av2.y;
      // B fragment: B[k][c] = x[b][nc+kp][c0+lane15] from LDS
      v2f bb;
      bb.x = xs[kp * 16 + lane15];
      bb.y = xs[(kp + 1) * 16 + lane15];
      acc = __builtin_amdgcn_wmma_f32_16x16x4_f32(
          false, a, false, bb, (short)0, acc, false, false);
    }
  }

  // D: rows = m, cols = c. lanes0-15: m=m0+v; lanes16-31: m=m0+v+8.
  const int mrow = m0 + ((lane >> 4) << 3);
  float* ob = out + ((size_t)b * MDIM + mrow) * CDIM + c0 + lane15;
#pragma unroll
  for (int v = 0; v < 8; ++v)
    ob[(size_t)v * CDIM] = acc[v];
}

// ---------------------------------------------------------------------------
extern "C" void kernel_launch(void* const* d_in, const int* in_sizes, int n_in,
                              void* d_out, int out_size, void* d_ws,
                              size_t ws_size, hipStream_t stream) {
  const float* x = (const float*)d_in[0];     // (8, 4096, 1024)
  const float* W = (const float*)d_in[1];     // (512, 1024)
  const float* bias = (const float*)d_in[2];  // (512,)
  const float* av = (const float*)d_in[3];    // (64, 512)
  float* out = (float*)d_out;                 // (8, 64, 1024)

  // Workspace layout
  float* Vt2 = (float*)d_ws;                        // 1024*64 floats (pairs)
  float* beta = Vt2 + (size_t)CDIM * MDIM;          // 64 floats
  float2* logitsP = (float2*)(beta + MDIM);         // 8*2048*64 float2
  (void)in_sizes; (void)n_in; (void)out_size; (void)ws_size;

  ap_fold_kernel<<<dim3(256), dim3(256), 0, stream>>>(W, bias, av, Vt2, beta);
  ap_logits_kernel<<<dim3(NDIM / 16, BDIM), dim3(128), 0, stream>>>(
      x, (const float2*)Vt2, beta, logitsP);
  ap_softmax_kernel<<<dim3(BDIM), dim3(512), 0, stream>>>(logitsP);
  ap_pool_kernel<<<dim3(CDIM / 16, BDIM), dim3(128), 0, stream>>>(
      x, logitsP, out);
}